// DGCN_37967510896844
// MI455X (gfx1250) — compile-verified
//
#include <hip/hip_runtime.h>

typedef float v2f __attribute__((ext_vector_type(2)));
typedef float v8f __attribute__((ext_vector_type(8)));

#define B_ 8
#define F_ 2048
#define L_ 1024
#define THRESH_ 0.81f
#define BN_EPS_ 1e-5f

// ---------------------------------------------------------------------------
// Kernel 1: h_raw[b,f] = mean_l( prelu(x[b,f,l], w1) )
// One block per (b,f) row; 256 threads x float4 = 1024 elements, coalesced.
// ---------------------------------------------------------------------------
__global__ void dgcn_reduce(const float* __restrict__ x,
                            const float* __restrict__ prelu1_w,
                            float* __restrict__ h_raw) {
    const int row = blockIdx.x;            // b*F + f
    const int tid = threadIdx.x;           // 0..255
    const float pw = prelu1_w[0];

    const float4 v =
        reinterpret_cast<const float4*>(x + (size_t)row * L_)[tid];

    float s = 0.0f;
    s += (v.x >= 0.0f) ? v.x : pw * v.x;
    s += (v.y >= 0.0f) ? v.y : pw * v.y;
    s += (v.z >= 0.0f) ? v.z : pw * v.z;
    s += (v.w >= 0.0f) ? v.w : pw * v.w;

    __shared__ float red[256];
    red[tid] = s;
    __syncthreads();
#pragma unroll
    for (int off = 128; off > 0; off >>= 1) {
        if (tid < off) red[tid] += red[tid + off];
        __syncthreads();
    }
    if (tid == 0) h_raw[row] = red[0] * (1.0f / (float)L_);
}

// ---------------------------------------------------------------------------
// Kernel 2: training-mode BatchNorm across the batch (per channel f, B=8).
// ---------------------------------------------------------------------------
__global__ void dgcn_bn(const float* __restrict__ h_raw,
                        const float* __restrict__ gamma,
                        const float* __restrict__ beta,
                        float* __restrict__ h) {
    const int f = blockIdx.x * blockDim.x + threadIdx.x;
    if (f >= F_) return;

    float v[B_];
    float mu = 0.0f;
#pragma unroll
    for (int b = 0; b < B_; ++b) {
        v[b] = h_raw[b * F_ + f];
        mu += v[b];
    }
    mu *= (1.0f / (float)B_);
    float var = 0.0f;
#pragma unroll
    for (int b = 0; b < B_; ++b) {
        const float d = v[b] - mu;
        var += d * d;
    }
    var *= (1.0f / (float)B_);           // biased variance
    const float sc = gamma[f] * rsqrtf(var + BN_EPS_);
    const float bs = beta[f];
#pragma unroll
    for (int b = 0; b < B_; ++b)
        h[b * F_ + f] = (v[b] - mu) * sc + bs;
}

// ---------------------------------------------------------------------------
// Kernel 3: degree + D^-1/2.
// deg[b,n] = 1 + #{ m : h[b,n]*h[b,m] > THRESH }   (the +1 is the identity)
// One block handles 256 rows of one batch; h[b,:] staged in LDS (8 KB).
// ---------------------------------------------------------------------------
__global__ void dgcn_deg(const float* __restrict__ h,
                         float* __restrict__ dinv) {
    __shared__ float sh[F_];
    const int b   = blockIdx.x >> 3;
    const int g   = blockIdx.x & 7;
    const int tid = threadIdx.x;
    const float* hb = h + b * F_;

#pragma unroll
    for (int k = 0; k < F_; k += 256) sh[k + tid] = hb[k + tid];
    __syncthreads();

    const int n = g * 256 + tid;
    const float hn = sh[n];
    int cnt = 1;                                  // identity term
    for (int m = 0; m < F_; ++m)
        cnt += (hn * sh[m] > THRESH_) ? 1 : 0;    // LDS broadcast reads
    dinv[b * F_ + n] = rsqrtf((float)cnt);
}

// ---------------------------------------------------------------------------
// Kernel 4: one wave per 16x16 output tile.
// sim tile via V_WMMA_F32_16X16X4_F32 with only K=0 populated:
//   A[i,0] = h[b, tn*16+i], B[0,j] = h[b, tm*16+j]  ->  D[i,j] = h_n * h_m.
// Epilogue: threshold + identity + d_n*d_m scaling + output PReLU, store.
// C/D layout: VGPR r, lane l -> row = r + 8*(l>>4), col = l&15.
// A (16x4 f32) layout: lanes 0-15 hold K=0 (VGPR0); we zero everything else.
// ---------------------------------------------------------------------------
__global__ void dgcn_out(const float* __restrict__ h,
                         const float* __restrict__ dinv,
                         const float* __restrict__ prelu2_w,
                         float* __restrict__ out) {
    const int tid  = threadIdx.x;
    const int wave = tid >> 5;
    const int lane = tid & 31;

    const unsigned tile = blockIdx.x * 8u + (unsigned)wave; // 0..131071
    const int b          = (int)(tile >> 14);               // / (128*128)
    const unsigned rem   = tile & 16383u;
    const int tn         = (int)(rem >> 7);                 // tile row
    const int tm         = (int)(rem & 127u);               // tile col

    const float* hb = h    + b * F_;
    const float* db = dinv + b * F_;

    const int j  = lane & 15;
    const int hi = lane >> 4;

    // Only K=0 carries data; lanes 16-31 (K=2) and second VGPR (K=1/3) are 0.
    const float av = (lane < 16) ? hb[tn * 16 + j] : 0.0f;
    const float bv = (lane < 16) ? hb[tm * 16 + j] : 0.0f;
    v2f A  = {av, 0.0f};
    v2f Bm = {bv, 0.0f};
    v8f C  = {};
    v8f D  = __builtin_amdgcn_wmma_f32_16x16x4_f32(
        /*neg_a=*/false, A, /*neg_b=*/false, Bm,
        /*c_mod=*/(short)0, C, /*reuse_a=*/false, /*reuse_b=*/false);

    const int   m  = tm * 16 + j;
    const float dm = db[m];
    const float w2 = prelu2_w[0];
    float* o = out + (size_t)b * (size_t)F_ * (size_t)F_ + (size_t)m;

#pragma unroll
    for (int r = 0; r < 8; ++r) {
        const int n = tn * 16 + r + 8 * hi;
        float a = (D[r] > THRESH_) ? 1.0f : 0.0f;
        a += (n == m) ? 1.0f : 0.0f;
        float val = db[n] * dm * a;
        val = (val >= 0.0f) ? val : w2 * val;   // output PReLU
        o[(size_t)n * (size_t)F_] = val;
    }
}

// ---------------------------------------------------------------------------
extern "C" void kernel_launch(void* const* d_in, const int* in_sizes, int n_in,
                              void* d_out, int out_size, void* d_ws, size_t ws_size,
                              hipStream_t stream) {
    (void)in_sizes; (void)n_in; (void)out_size; (void)ws_size;

    const float* x        = (const float*)d_in[0];  // [8,2048,1024]
    const float* prelu1_w = (const float*)d_in[1];  // [1]
    const float* prelu2_w = (const float*)d_in[2];  // [1]
    const float* bn_w     = (const float*)d_in[3];  // [2048]
    const float* bn_b     = (const float*)d_in[4];  // [2048]
    float* out            = (float*)d_out;          // [8,2048,2048]

    float* h_raw = (float*)d_ws;                    // 16384 floats
    float* hbuf  = h_raw + B_ * F_;                 // 16384 floats
    float* dinv  = hbuf  + B_ * F_;                 // 16384 floats

    // 1) PReLU + mean over L
    dgcn_reduce<<<B_ * F_, 256, 0, stream>>>(x, prelu1_w, h_raw);
    // 2) BatchNorm (batch statistics)
    dgcn_bn<<<F_ / 256, 256, 0, stream>>>(h_raw, bn_w, bn_b, hbuf);
    // 3) degrees -> D^-1/2
    dgcn_deg<<<B_ * (F_ / 256), 256, 0, stream>>>(hbuf, dinv);
    // 4) WMMA outer-product tiles + normalize + PReLU + store (BW-bound)
    dgcn_out<<<(B_ * 128 * 128) / 8, 256, 0, stream>>>(hbuf, dinv, prelu2_w, out);
}